// KAM_62079457296519
// MI455X (gfx1250) — compile-verified
//
#include <hip/hip_runtime.h>

// ---------------------------------------------------------------------------
// B=16, C=64, H=W=192, K=3. All GEMM stages use V_WMMA_F32_16X16X4_F32
// (fp32 precision on the matrix pipe). K-dim is khw-major / channel-minor
// so 3x3 offsets are compile-time constants and per-K-step address math is
// one constant-stride add. Each wave computes a 32(M) x 64(N) tile:
// 2 A loads + 8 B loads feed 8 WMMAs per K-step (8 independent acc chains).
// ---------------------------------------------------------------------------

typedef __attribute__((ext_vector_type(2))) float v2f;
typedef __attribute__((ext_vector_type(8))) float v8f;

__device__ __forceinline__ v8f wmma4(v2f a, v2f b, v8f c) {
  return __builtin_amdgcn_wmma_f32_16x16x4_f32(
      false, a, false, b, (short)0, c, false, false);
}

// --------------------------------------------------------------------------
// Tiny one-time weight transpose: in[r][ic][khw] -> out[r][khw][ic]
// --------------------------------------------------------------------------
__global__ void transpose_k9(const float* __restrict__ in, float* __restrict__ out) {
  const int r = blockIdx.x;
  const int j = threadIdx.x;                 // j = ic*9 + khw
  const int ic = j / 9, khw = j - ic * 9;
  out[(size_t)r * 576 + khw * 64 + ic] = in[(size_t)r * 576 + j];
}

// --------------------------------------------------------------------------
// Stage 1: qk = conv(x, w_qk, stride 3, valid). GEMM M=128, N=4096, K=576.
// --------------------------------------------------------------------------
__global__ void conv_qk_wmma(const float* __restrict__ x,
                             const float* __restrict__ wqkt,
                             float* __restrict__ qk) {
  const int b  = blockIdx.z;
  const int m0 = blockIdx.y << 5;            // 32 oc per wave
  const int n0 = blockIdx.x << 6;            // 64 pixels per wave
  const int lane = threadIdx.x & 31, lrow = lane & 15, khalf = lane >> 4;

  const float* Arow0 = wqkt + (size_t)(m0 + lrow) * 576 + (khalf << 1);
  const float* Arow1 = Arow0 + 16 * 576;
  const float* xb = x + (size_t)b * 64 * 192 * 192;

  int py[4], px[4];
#pragma unroll
  for (int s = 0; s < 4; ++s) {
    const int p = n0 + (s << 4) + lrow;
    py[s] = p >> 6; px[s] = p & 63;
  }

  v8f acc[2][4] = {};
#pragma unroll
  for (int khw = 0; khw < 9; ++khw) {
    const int kh = khw / 3, kw = khw - kh * 3;
    const float* Ak0 = Arow0 + khw * 64;
    const float* Ak1 = Arow1 + khw * 64;
    const float* bb[4];
#pragma unroll
    for (int s = 0; s < 4; ++s)
      bb[s] = xb + (size_t)(3 * py[s] + kh) * 192 + (3 * px[s] + kw)
                 + (size_t)(khalf << 1) * 36864;
#pragma unroll 2
    for (int cs = 0; cs < 16; ++cs) {
      const size_t c4 = (size_t)(cs << 2);
      v2f a0 = *(const v2f*)(Ak0 + c4);
      v2f a1 = *(const v2f*)(Ak1 + c4);
#pragma unroll
      for (int s = 0; s < 4; ++s) {
        v2f bv = {bb[s][c4 * 36864], bb[s][(c4 + 1) * 36864]};
        acc[0][s] = wmma4(a0, bv, acc[0][s]);
        acc[1][s] = wmma4(a1, bv, acc[1][s]);
      }
    }
  }
#pragma unroll
  for (int mi = 0; mi < 2; ++mi)
#pragma unroll
    for (int s = 0; s < 4; ++s) {
      float* outp = qk + ((size_t)b * 128 + m0 + mi * 16 + khalf * 8) * 4096
                       + n0 + (s << 4) + lrow;
#pragma unroll
      for (int r = 0; r < 8; ++r) outp[(size_t)r * 4096] = acc[mi][s][r];
    }
}

// --------------------------------------------------------------------------
// Stage 2: per-sample L2 norms of q (ch 0..63) and k (ch 64..127).
// --------------------------------------------------------------------------
__global__ void qk_norms(const float* __restrict__ qk, float* __restrict__ norms) {
  const int which = blockIdx.x & 1;
  const int b     = blockIdx.x >> 1;
  const float* base = qk + ((size_t)b * 128 + which * 64) * 4096;
  float s = 0.f;
  for (int i = threadIdx.x; i < 64 * 4096; i += blockDim.x) {
    float v = base[i]; s += v * v;
  }
  __shared__ float red[256];
  red[threadIdx.x] = s; __syncthreads();
  for (int off = 128; off > 0; off >>= 1) {
    if ((int)threadIdx.x < off) red[threadIdx.x] += red[threadIdx.x + off];
    __syncthreads();
  }
  if (threadIdx.x == 0) norms[b * 2 + which] = fmaxf(sqrtf(red[0]), 1e-12f);
}

// --------------------------------------------------------------------------
// Stage 3: attn[b,d,c,i,j] = <k[b,d], q[b,c] shifted (i-2,j-2)> / (nq*nk)
// GEMM per (b,shift): M=64 (d, 32 per wave), N=64 (c, full strip), K=4096.
// Pad rows skip their WMMAs entirely (wave-uniform branch).
// --------------------------------------------------------------------------
__global__ void attn_wmma(const float* __restrict__ qk,
                          const float* __restrict__ norms,
                          float* __restrict__ attn) {
  const int b  = blockIdx.z;
  const int ij = blockIdx.y;
  const int m0 = blockIdx.x << 5;            // d tile (32 per wave)
  const int di = ij / 5 - 2;
  const int dj = (ij - (ij / 5) * 5) - 2;
  const int lane = threadIdx.x & 31, lrow = lane & 15, khalf = lane >> 4;

  const float* Arow0 = qk + ((size_t)b * 128 + 64 + m0 + lrow) * 4096 + (khalf << 1);
  const float* Arow1 = Arow0 + (size_t)16 * 4096;
  const float* qr[4];
#pragma unroll
  for (int s = 0; s < 4; ++s)
    qr[s] = qk + ((size_t)b * 128 + (s << 4) + lrow) * 4096;

  v8f acc[2][4] = {};
  for (int h = 0; h < 64; ++h) {
    const int hh = h + di;
    if (hh < 0 || hh >= 64) continue;        // zero row: skip (uniform)
    const float* Ah0 = Arow0 + (h << 6);
    const float* Ah1 = Arow1 + (h << 6);
    const int qbase = (hh << 6) + dj + (khalf << 1);
#pragma unroll 2
    for (int wc = 0; wc < 16; ++wc) {
      const int w4 = wc << 2;
      v2f a0 = *(const v2f*)(Ah0 + w4);
      v2f a1 = *(const v2f*)(Ah1 + w4);
      const int ww0 = w4 + dj + (khalf << 1);
      const bool ok0 = (ww0 >= 0) && (ww0 < 64);
      const bool ok1 = (ww0 + 1 >= 0) && (ww0 + 1 < 64);
      const int o0 = qbase + w4, o1 = o0 + 1;
#pragma unroll
      for (int s = 0; s < 4; ++s) {
        v2f bv = {ok0 ? qr[s][o0] : 0.f, ok1 ? qr[s][o1] : 0.f};
        acc[0][s] = wmma4(a0, bv, acc[0][s]);
        acc[1][s] = wmma4(a1, bv, acc[1][s]);
      }
    }
  }
  const float scale = 1.f / (norms[b * 2 + 0] * norms[b * 2 + 1]);
#pragma unroll
  for (int mi = 0; mi < 2; ++mi)
#pragma unroll
    for (int s = 0; s < 4; ++s) {
      const int c = (s << 4) + lrow;
#pragma unroll
      for (int r = 0; r < 8; ++r) {
        const int d = m0 + mi * 16 + khalf * 8 + r;
        attn[(((size_t)b * 64 + d) * 64 + c) * 25 + ij] = acc[mi][s][r] * scale;
      }
    }
}

// --------------------------------------------------------------------------
// Stage 4: ak_conv[b,cb,oc,yx] = valid 3x3 conv of attn[b,cb] with weight.
// GEMM per b: M=64 (oc, 32 per wave), N=576 (cb*9+yx), K=576 (khw-major).
// --------------------------------------------------------------------------
__global__ void ak_conv_wmma(const float* __restrict__ attn,
                             const float* __restrict__ wgtt,
                             float* __restrict__ akc) {
  const int b  = blockIdx.z;
  const int m0 = blockIdx.y << 5;
  const int n0 = blockIdx.x << 6;
  const int lane = threadIdx.x & 31, lrow = lane & 15, khalf = lane >> 4;

  const float* Arow0 = wgtt + (size_t)(m0 + lrow) * 576 + (khalf << 1);
  const float* Arow1 = Arow0 + 16 * 576;
  int cb[4], yx[4], oy[4], ox[4];
#pragma unroll
  for (int s = 0; s < 4; ++s) {
    const int n = n0 + (s << 4) + lrow;
    cb[s] = n / 9; yx[s] = n - cb[s] * 9;
    oy[s] = yx[s] / 3; ox[s] = yx[s] - oy[s] * 3;
  }

  v8f acc[2][4] = {};
#pragma unroll
  for (int khw = 0; khw < 9; ++khw) {
    const int kh = khw / 3, kw = khw - kh * 3;
    const float* Ak0 = Arow0 + khw * 64;
    const float* Ak1 = Arow1 + khw * 64;
    const float* bb[4];
#pragma unroll
    for (int s = 0; s < 4; ++s)
      bb[s] = attn + ((size_t)b * 64 + cb[s]) * 1600
                   + (oy[s] + kh) * 5 + (ox[s] + kw) + (size_t)(khalf << 1) * 25;
#pragma unroll 2
    for (int cs = 0; cs < 16; ++cs) {
      const size_t c4 = (size_t)(cs << 2);
      v2f a0 = *(const v2f*)(Ak0 + c4);
      v2f a1 = *(const v2f*)(Ak1 + c4);
#pragma unroll
      for (int s = 0; s < 4; ++s) {
        v2f bv = {bb[s][c4 * 25], bb[s][(c4 + 1) * 25]};
        acc[0][s] = wmma4(a0, bv, acc[0][s]);
        acc[1][s] = wmma4(a1, bv, acc[1][s]);
      }
    }
  }
#pragma unroll
  for (int mi = 0; mi < 2; ++mi)
#pragma unroll
    for (int s = 0; s < 4; ++s)
#pragma unroll
      for (int r = 0; r < 8; ++r) {
        const int oc = m0 + mi * 16 + khalf * 8 + r;
        akc[(((size_t)b * 64 + cb[s]) * 64 + oc) * 9 + yx[s]] = acc[mi][s][r];
      }
}

// --------------------------------------------------------------------------
// Stage 5: per (b,oc) norm over (d,khw); emit kern in khw-major layout:
// kernt[b][oc][khw][ci] = wgtt + akc(transposed C dims) * T / max(norm,eps)
// --------------------------------------------------------------------------
__global__ void build_kern(const float* __restrict__ akc,
                           const float* __restrict__ wgtt,
                           const float* __restrict__ temperature,
                           float* __restrict__ kernt) {
  const int b  = blockIdx.y;
  const int oc = blockIdx.x;
  float s = 0.f;
  for (int j = threadIdx.x; j < 576; j += blockDim.x) {   // j = khw*64 + d
    const int d = j & 63, khw = j >> 6;
    const float v = akc[(((size_t)b * 64 + d) * 64 + oc) * 9 + khw];
    s += v * v;
  }
  __shared__ float red[256];
  red[threadIdx.x] = s; __syncthreads();
  for (int off = 128; off > 0; off >>= 1) {
    if ((int)threadIdx.x < off) red[threadIdx.x] += red[threadIdx.x + off];
    __syncthreads();
  }
  const float scale = temperature[0] / fmaxf(sqrtf(red[0]), 1e-12f);
  for (int j = threadIdx.x; j < 576; j += blockDim.x) {
    const int d = j & 63, khw = j >> 6;
    const float v = akc[(((size_t)b * 64 + d) * 64 + oc) * 9 + khw];
    kernt[((size_t)b * 64 + oc) * 576 + j] = wgtt[(size_t)oc * 576 + j] + v * scale;
  }
}

// --------------------------------------------------------------------------
// Stage 6 (dominant, ~43.5 GFLOP): per-sample 3x3 conv, pad 1, relu.
// GEMM per b: M=64 (co, 32 per wave), N=36864 (pixels), K=576 (khw-major).
// 64-pixel strip sits in one image row (192 = 3*64) -> row predicate is
// wave-uniform (skip pad rows); only column edges need per-lane selects.
// --------------------------------------------------------------------------
__global__ void final_conv_wmma(const float* __restrict__ x,
                                const float* __restrict__ kernt,
                                float* __restrict__ outp) {
  const int b  = blockIdx.z;
  const int m0 = blockIdx.y << 5;
  const int n0 = blockIdx.x << 6;
  const int lane = threadIdx.x & 31, lrow = lane & 15, khalf = lane >> 4;

  const float* Arow0 = kernt + ((size_t)b * 64 + m0 + lrow) * 576 + (khalf << 1);
  const float* Arow1 = Arow0 + 16 * 576;
  const int py  = n0 / 192;                  // uniform over the wave
  const int px0 = n0 - py * 192;
  const float* xb = x + (size_t)b * 64 * 36864;

  v8f acc[2][4] = {};
#pragma unroll
  for (int khw = 0; khw < 9; ++khw) {
    const int kh = khw / 3, kw = khw - kh * 3;
    const int iy = py + kh - 1;
    if (iy < 0 || iy >= 192) continue;       // pad row: skip (uniform)
    const float* Ak0 = Arow0 + khw * 64;
    const float* Ak1 = Arow1 + khw * 64;
    const float* bb[4];
    bool ok[4];
#pragma unroll
    for (int s = 0; s < 4; ++s) {
      const int ix = px0 + (s << 4) + lrow + kw - 1;
      ok[s] = (ix >= 0) && (ix < 192);
      bb[s] = xb + (size_t)iy * 192 + ix + (size_t)(khalf << 1) * 36864;
    }
#pragma unroll 2
    for (int cs = 0; cs < 16; ++cs) {
      const size_t c4 = (size_t)(cs << 2);
      v2f a0 = *(const v2f*)(Ak0 + c4);
      v2f a1 = *(const v2f*)(Ak1 + c4);
#pragma unroll
      for (int s = 0; s < 4; ++s) {
        v2f bv = {ok[s] ? bb[s][c4 * 36864] : 0.f,
                  ok[s] ? bb[s][(c4 + 1) * 36864] : 0.f};
        acc[0][s] = wmma4(a0, bv, acc[0][s]);
        acc[1][s] = wmma4(a1, bv, acc[1][s]);
      }
    }
  }
#pragma unroll
  for (int mi = 0; mi < 2; ++mi)
#pragma unroll
    for (int s = 0; s < 4; ++s)
#pragma unroll
      for (int r = 0; r < 8; ++r) {
        const int co = m0 + mi * 16 + khalf * 8 + r;
        outp[((size_t)b * 64 + co) * 36864 + n0 + (s << 4) + lrow] =
            fmaxf(acc[mi][s][r], 0.f);
      }
}

// --------------------------------------------------------------------------
extern "C" void kernel_launch(void* const* d_in, const int* in_sizes, int n_in,
                              void* d_out, int out_size, void* d_ws, size_t ws_size,
                              hipStream_t stream) {
  (void)in_sizes; (void)n_in; (void)out_size; (void)ws_size;
  const float* x    = (const float*)d_in[0];  // (16,64,192,192)
  const float* wqk  = (const float*)d_in[1];  // (128,64,3,3)
  const float* wgt  = (const float*)d_in[2];  // (64,64,3,3)
  const float* temp = (const float*)d_in[3];  // scalar
  float* out = (float*)d_out;                 // (16,64,192,192)

  // workspace (floats): qk | norms | attn | akc | kernt | wqkt | wgtt (~45.3MB)
  float* ws    = (float*)d_ws;
  float* qk    = ws;                                  // 16*128*4096
  float* norms = qk + (size_t)16 * 128 * 4096;        // 32 (pad 64)
  float* attn  = norms + 64;                          // 16*64*64*25
  float* akc   = attn + (size_t)16 * 64 * 64 * 25;    // 16*64*64*9
  float* kernt = akc + (size_t)16 * 64 * 64 * 9;      // 16*64*64*9
  float* wqkt  = kernt + (size_t)16 * 64 * 64 * 9;    // 128*576
  float* wgtt  = wqkt + (size_t)128 * 576;            // 64*576

  transpose_k9   <<<dim3(128),          576, 0, stream>>>(wqk, wqkt);
  transpose_k9   <<<dim3(64),           576, 0, stream>>>(wgt, wgtt);
  conv_qk_wmma   <<<dim3(64, 4, 16),     32, 0, stream>>>(x, wqkt, qk);
  qk_norms       <<<dim3(32),           256, 0, stream>>>(qk, norms);
  attn_wmma      <<<dim3(2, 25, 16),     32, 0, stream>>>(qk, norms, attn);
  ak_conv_wmma   <<<dim3(9, 2, 16),      32, 0, stream>>>(attn, wgtt, akc);
  build_kern     <<<dim3(64, 16),       256, 0, stream>>>(akc, wgtt, temp, kernt);
  final_conv_wmma<<<dim3(576, 2, 16),    32, 0, stream>>>(x, kernt, out);
}